// LuongAttnDecoderRNN_24550033063965
// MI455X (gfx1250) — compile-verified
//
#include <hip/hip_runtime.h>
#include <hip/hip_bf16.h>
#include <math.h>

// Problem constants (from reference)
#define Bc 64
#define Ec 1024
#define Hc 1024
#define Lc 2048
#define Vc 50257

typedef float v2f __attribute__((ext_vector_type(2)));
typedef float v8f __attribute__((ext_vector_type(8)));
typedef unsigned int u32x4 __attribute__((ext_vector_type(4)));
typedef int i32x4 __attribute__((ext_vector_type(4)));
typedef int i32x8 __attribute__((ext_vector_type(8)));

#if defined(__has_builtin)
#if __has_builtin(__builtin_amdgcn_tensor_load_to_lds) && __has_builtin(__builtin_amdgcn_s_wait_tensorcnt)
#define HAVE_TDM 1
#endif
#endif
#ifndef HAVE_TDM
#define HAVE_TDM 0
#endif

// ---------------------------------------------------------------------------
// WMMA GEMM:  C[64,N] = A[64,K] @ W[N,K]^T + bias[N]   (optional tanh)
//
// * One wave owns one 16-wide N tile and ALL four 16-row M tiles (M=64), so
//   every B fragment feeds 4 v_wmma_f32_16x16x4_f32 ops and the weight matrix
//   is streamed exactly once across the grid.
// * TDM path: per-wave double-buffered LDS staging of the 16xKC weight tile
//   via tensor_load_to_lds (2D D# descriptor), synchronized with
//   s_wait_tensorcnt; B fragments come from LDS (ds_load_b64).
// * A fragments (64xK activations, L2-resident) load direct from global and
//   are register double-buffered together with B so loads decouple from WMMAs.
// * Edge N tiles: clamp the tile base/valid-row-count for loads, predicate
//   only the stores -> EXEC is all-ones at every WMMA.
// ---------------------------------------------------------------------------
#define GEMM_WAVES 4
#define KC 64   // K-chunk staged per TDM transfer (multiple of 4; K % KC == 0)

#define WMMA4(af0, af1, af2, af3, bf)                                                            \
  c0 = __builtin_amdgcn_wmma_f32_16x16x4_f32(false, af0, false, bf, (short)0, c0, false, false); \
  c1 = __builtin_amdgcn_wmma_f32_16x16x4_f32(false, af1, false, bf, (short)0, c1, false, false); \
  c2 = __builtin_amdgcn_wmma_f32_16x16x4_f32(false, af2, false, bf, (short)0, c2, false, false); \
  c3 = __builtin_amdgcn_wmma_f32_16x16x4_f32(false, af3, false, bf, (short)0, c3, false, false);

template <int MODE> // 0 = none, 1 = tanh
__global__ __launch_bounds__(GEMM_WAVES * 32)
void wmma_gemm_nt(const float* __restrict__ A,   // [64, K] row-major
                  const float* __restrict__ W,   // [N,  K] row-major
                  const float* __restrict__ bias,// [N]
                  float* __restrict__ C,         // [64, N] row-major
                  int N, int K) {
  const int lane = threadIdx.x & 31;
  const int wave = __builtin_amdgcn_readfirstlane((int)(threadIdx.x >> 5));
  const int n0   = (blockIdx.x * GEMM_WAVES + wave) * 16;

  const int row = lane & 15;         // A: M row / B: N column within tile
  const int kk  = (lane >> 4) << 1;  // K sub-offset 0 or 2

  const float* __restrict__ a0 = A + (size_t)( 0 + row) * K + kk;
  const float* __restrict__ a1 = A + (size_t)(16 + row) * K + kk;
  const float* __restrict__ a2 = A + (size_t)(32 + row) * K + kk;
  const float* __restrict__ a3 = A + (size_t)(48 + row) * K + kk;

  v8f c0 = {}, c1 = {}, c2 = {}, c3 = {};

#if HAVE_TDM
  // ---- TDM-staged weight path -------------------------------------------
  __shared__ float smem[GEMM_WAVES * 2 * 16 * KC];
  float* __restrict__ buf0 = smem + (size_t)wave * (2 * 16 * KC);
  float* __restrict__ buf1 = buf0 + 16 * KC;

  // Valid rows of this tile; fully out-of-range tiles read tensor row 0.
  const int nv   = N - n0;
  const int rows = (nv >= 16) ? 16 : ((nv > 0) ? nv : 16);
  const size_t gbase = (size_t)(const void*)W +
                       (size_t)((nv > 0) ? n0 : 0) * (size_t)K * 4u;

  // D# group1 (constant over the K loop): data_size=4B, dims in elements.
  i32x8 g1;
  g1[0] = (int)(2u << 16);                                     // data_size=4B
  g1[1] = (int)(((unsigned)K & 0xFFFFu) << 16);                // tensor_dim0 lo
  g1[2] = (int)(((unsigned)K >> 16) | (((unsigned)N & 0xFFFFu) << 16));
  g1[3] = (int)(((unsigned)N >> 16) | ((unsigned)KC << 16));   // tile_dim0=KC
  g1[4] = rows;                                                // tile_dim1
  g1[5] = K;                                                   // dim0 stride lo
  g1[6] = 0;
  g1[7] = 0;
  const i32x4 gz4 = {0, 0, 0, 0};
  const i32x8 gz8 = {0, 0, 0, 0, 0, 0, 0, 0};

  // Prologue: stage chunk 0 into buf0.
  {
    const size_t ga = gbase;
    u32x4 g0;
    g0[0] = 1u;                                                // count=1
    g0[1] = (unsigned)(size_t)buf0;                            // LDS byte addr
    g0[2] = (unsigned)ga;
    g0[3] = (unsigned)((ga >> 32) & 0x1FFFFFFu) | (2u << 30);  // type=2
    __builtin_amdgcn_tensor_load_to_lds(g0, g1, gz4, gz4, gz8, 0);
  }

  int s = 0;
  for (int kc = 0; kc < K; kc += KC) {
    float* __restrict__ curb = s ? buf1 : buf0;
    float* __restrict__ nxtb = s ? buf0 : buf1;
    const bool more = (kc + KC) < K;
    if (more) {
      const size_t ga = gbase + (size_t)(kc + KC) * 4u;
      u32x4 g0;
      g0[0] = 1u;
      g0[1] = (unsigned)(size_t)nxtb;
      g0[2] = (unsigned)ga;
      g0[3] = (unsigned)((ga >> 32) & 0x1FFFFFFu) | (2u << 30);
      __builtin_amdgcn_tensor_load_to_lds(g0, g1, gz4, gz4, gz8, 0);
      __builtin_amdgcn_s_wait_tensorcnt((short)1);   // current chunk ready
    } else {
      __builtin_amdgcn_s_wait_tensorcnt((short)0);
    }

    // Register double-buffered fragment pipeline over this KC chunk.
    const float* __restrict__ bp = curb + (size_t)row * KC + kk;
    v2f bc = *(const v2f*)(bp);
    v2f x0 = *(const v2f*)(a0 + kc);
    v2f x1 = *(const v2f*)(a1 + kc);
    v2f x2 = *(const v2f*)(a2 + kc);
    v2f x3 = *(const v2f*)(a3 + kc);
#pragma unroll 4
    for (int k2 = 4; k2 < KC; k2 += 4) {
      v2f bn = *(const v2f*)(bp + k2);
      v2f y0 = *(const v2f*)(a0 + kc + k2);
      v2f y1 = *(const v2f*)(a1 + kc + k2);
      v2f y2 = *(const v2f*)(a2 + kc + k2);
      v2f y3 = *(const v2f*)(a3 + kc + k2);
      WMMA4(x0, x1, x2, x3, bc)
      bc = bn; x0 = y0; x1 = y1; x2 = y2; x3 = y3;
    }
    WMMA4(x0, x1, x2, x3, bc)
    s ^= 1;
  }
#else
  // ---- Fallback: direct NT global weight loads, register pipelined ------
  int wr = n0 + row; if (wr > N - 1) wr = N - 1;
  const float* __restrict__ wp = W + (size_t)wr * K + kk;

  v2f bc, x0, x1, x2, x3;
  bc.x = __builtin_nontemporal_load(wp);
  bc.y = __builtin_nontemporal_load(wp + 1);
  x0 = *(const v2f*)(a0); x1 = *(const v2f*)(a1);
  x2 = *(const v2f*)(a2); x3 = *(const v2f*)(a3);
#pragma unroll 4
  for (int k = 4; k < K; k += 4) {
    v2f bn;
    bn.x = __builtin_nontemporal_load(wp + k);
    bn.y = __builtin_nontemporal_load(wp + k + 1);
    v2f y0 = *(const v2f*)(a0 + k);
    v2f y1 = *(const v2f*)(a1 + k);
    v2f y2 = *(const v2f*)(a2 + k);
    v2f y3 = *(const v2f*)(a3 + k);
    WMMA4(x0, x1, x2, x3, bc)
    bc = bn; x0 = y0; x1 = y1; x2 = y2; x3 = y3;
  }
  WMMA4(x0, x1, x2, x3, bc)
#endif

  // D layout: lane(0-15)=col n, VGPR i = row i (lanes 16-31: rows 8-15)
  const int col = lane & 15;
  const int rb  = (lane >> 4) << 3; // 0 or 8
  const int n   = n0 + col;
  if (n < N) {
    const float bv = bias[n];
#pragma unroll
    for (int i = 0; i < 8; ++i) {
      float v;
      v = c0[i] + bv; if (MODE) v = tanhf(v); C[(size_t)( 0 + rb + i) * N + n] = v;
      v = c1[i] + bv; if (MODE) v = tanhf(v); C[(size_t)(16 + rb + i) * N + n] = v;
      v = c2[i] + bv; if (MODE) v = tanhf(v); C[(size_t)(32 + rb + i) * N + n] = v;
      v = c3[i] + bv; if (MODE) v = tanhf(v); C[(size_t)(48 + rb + i) * N + n] = v;
    }
  }
}

// ---------------------------------------------------------------------------
// Embedding gather: emb[b,:] = embedding[input_seq[b], :]
// ---------------------------------------------------------------------------
__global__ void gather_embed(const int* __restrict__ idx,
                             const float* __restrict__ table,
                             float* __restrict__ emb) {
  const int b = blockIdx.x;
  const size_t r = (size_t)idx[b] * Ec;
  for (int e = threadIdx.x; e < Ec; e += blockDim.x)
    emb[(size_t)b * Ec + e] = table[r + e];
}

// ---------------------------------------------------------------------------
// GRU pointwise gates: h = (1-z)*n + z*h_prev
// ---------------------------------------------------------------------------
__global__ void gru_gate(const float* __restrict__ gi,
                         const float* __restrict__ gh,
                         const float* __restrict__ hprev,
                         float* __restrict__ hout,
                         float* __restrict__ hid_out) {
  const int b = blockIdx.x;
  const size_t g = (size_t)b * 3 * Hc;
  for (int j = threadIdx.x; j < Hc; j += blockDim.x) {
    const float ir = gi[g + j],          hr = gh[g + j];
    const float iz = gi[g + Hc + j],     hz = gh[g + Hc + j];
    const float in_ = gi[g + 2*Hc + j],  hn = gh[g + 2*Hc + j];
    const float r = 1.f / (1.f + expf(-(ir + hr)));
    const float z = 1.f / (1.f + expf(-(iz + hz)));
    const float n = tanhf(in_ + r * hn);
    const float h = (1.f - z) * n + z * hprev[(size_t)b * Hc + j];
    hout[(size_t)b * Hc + j]    = h;
    hid_out[(size_t)b * Hc + j] = h;
  }
}

// ---------------------------------------------------------------------------
// scores[b,l] = <h1[b,:], enc[l,b,:]>  (one wave per (b,l), NT stream of enc)
// ---------------------------------------------------------------------------
__global__ void attn_scores(const float* __restrict__ h1,
                            const float* __restrict__ enc,
                            float* __restrict__ scores) {
  const int b    = blockIdx.y;
  const int l    = blockIdx.x * 8 + (threadIdx.x >> 5);
  const int lane = threadIdx.x & 31;
  const float* __restrict__ e  = enc + ((size_t)l * Bc + b) * Hc;
  const float* __restrict__ hv = h1 + (size_t)b * Hc;
  float s = 0.f;
  for (int h = lane; h < Hc; h += 32)
    s += hv[h] * __builtin_nontemporal_load(e + h);
  for (int off = 16; off; off >>= 1) s += __shfl_xor(s, off, 32);
  if (lane == 0) scores[(size_t)b * Lc + l] = s;
}

// ---------------------------------------------------------------------------
// In-place log_softmax over L per batch row; also copies to attn output.
// ---------------------------------------------------------------------------
__global__ void log_softmax_k(float* __restrict__ s, float* __restrict__ attn_out) {
  __shared__ float red[8];
  __shared__ float bcast;
  const int b = blockIdx.x, t = threadIdx.x, lane = t & 31, w = t >> 5;
  const size_t base = (size_t)b * Lc;

  float m = -3.402823466e38f;
  for (int l = t; l < Lc; l += 256) m = fmaxf(m, s[base + l]);
  for (int off = 16; off; off >>= 1) m = fmaxf(m, __shfl_xor(m, off, 32));
  if (lane == 0) red[w] = m;
  __syncthreads();
  if (t == 0) { float v = red[0]; for (int i = 1; i < 8; ++i) v = fmaxf(v, red[i]); bcast = v; }
  __syncthreads();
  const float mx = bcast;

  float sum = 0.f;
  for (int l = t; l < Lc; l += 256) sum += expf(s[base + l] - mx);
  for (int off = 16; off; off >>= 1) sum += __shfl_xor(sum, off, 32);
  if (lane == 0) red[w] = sum;
  __syncthreads();
  if (t == 0) { float v = 0.f; for (int i = 0; i < 8; ++i) v += red[i]; bcast = logf(v); }
  __syncthreads();
  const float lse = mx + bcast;

  for (int l = t; l < Lc; l += 256) {
    const float v = s[base + l] - lse;
    s[base + l] = v;
    attn_out[base + l] = v;
  }
}

// ---------------------------------------------------------------------------
// context[b,h] = sum_l w[b,l] * enc[l,b,h]   (one thread per h, deterministic)
// ---------------------------------------------------------------------------
__global__ __launch_bounds__(1024)
void context_k(const float* __restrict__ logp,
               const float* __restrict__ enc,
               float* __restrict__ ctx) {
  const int b = blockIdx.x, h = threadIdx.x;
  const float* __restrict__ e = enc + (size_t)b * Hc + h;
  const float* __restrict__ w = logp + (size_t)b * Lc;
  float acc = 0.f;
  for (int l = 0; l < Lc; ++l)
    acc += w[l] * __builtin_nontemporal_load(e + (size_t)l * Bc * Hc);
  ctx[(size_t)b * Hc + h] = acc;
}

// ---------------------------------------------------------------------------
// cc[b,:] = [h1[b,:], context[b,:]]
// ---------------------------------------------------------------------------
__global__ void concat_build(const float* __restrict__ h1,
                             const float* __restrict__ ctx,
                             float* __restrict__ cc) {
  const int b = blockIdx.x;
  for (int j = threadIdx.x; j < 2 * Hc; j += blockDim.x)
    cc[(size_t)b * 2 * Hc + j] = (j < Hc) ? h1[(size_t)b * Hc + j]
                                          : ctx[(size_t)b * Hc + (j - Hc)];
}

// ---------------------------------------------------------------------------
extern "C" void kernel_launch(void* const* d_in, const int* in_sizes, int n_in,
                              void* d_out, int out_size, void* d_ws, size_t ws_size,
                              hipStream_t stream) {
  const int*   input_seq   = (const int*)  d_in[0];
  const float* last_hidden = (const float*)d_in[1];   // [2,B,H]
  const float* enc         = (const float*)d_in[2];   // [L,B,H]
  const float* embedding   = (const float*)d_in[3];   // [V,E]
  const float* W_ih0 = (const float*)d_in[4];
  const float* W_hh0 = (const float*)d_in[5];
  const float* b_ih0 = (const float*)d_in[6];
  const float* b_hh0 = (const float*)d_in[7];
  const float* W_ih1 = (const float*)d_in[8];
  const float* W_hh1 = (const float*)d_in[9];
  const float* b_ih1 = (const float*)d_in[10];
  const float* b_hh1 = (const float*)d_in[11];
  const float* concat_W = (const float*)d_in[12];
  const float* concat_b = (const float*)d_in[13];
  const float* out_W    = (const float*)d_in[14];
  const float* out_b    = (const float*)d_in[15];

  float* out  = (float*)d_out;                 // [B,V]
  float* hid0 = out + (size_t)Bc * Vc;         // [B,H]
  float* hid1 = hid0 + (size_t)Bc * Hc;        // [B,H]
  float* attn = hid0 + (size_t)2 * Bc * Hc;    // [B,1,L]

  float* ws = (float*)d_ws;
  size_t o = 0;
  float* emb  = ws + o; o += (size_t)Bc * Ec;
  float* gi   = ws + o; o += (size_t)Bc * 3 * Hc;
  float* gh   = ws + o; o += (size_t)Bc * 3 * Hc;
  float* h0   = ws + o; o += (size_t)Bc * Hc;
  float* h1   = ws + o; o += (size_t)Bc * Hc;
  float* logp = ws + o; o += (size_t)Bc * Lc;
  float* ctx  = ws + o; o += (size_t)Bc * Hc;
  float* cc   = ws + o; o += (size_t)Bc * 2 * Hc;
  float* cout = ws + o; o += (size_t)Bc * Hc;

  const int gemm_threads = GEMM_WAVES * 32;

  // 1. embedding gather
  gather_embed<<<Bc, 256, 0, stream>>>(input_seq, embedding, emb);

  // 2. GRU layer 0
  wmma_gemm_nt<0><<<(3 * Hc + 63) / 64, gemm_threads, 0, stream>>>(emb, W_ih0, b_ih0, gi, 3 * Hc, Ec);
  wmma_gemm_nt<0><<<(3 * Hc + 63) / 64, gemm_threads, 0, stream>>>(last_hidden, W_hh0, b_hh0, gh, 3 * Hc, Hc);
  gru_gate<<<Bc, 256, 0, stream>>>(gi, gh, last_hidden, h0, hid0);

  // 3. GRU layer 1
  wmma_gemm_nt<0><<<(3 * Hc + 63) / 64, gemm_threads, 0, stream>>>(h0, W_ih1, b_ih1, gi, 3 * Hc, Hc);
  wmma_gemm_nt<0><<<(3 * Hc + 63) / 64, gemm_threads, 0, stream>>>(last_hidden + (size_t)Bc * Hc, W_hh1, b_hh1, gh, 3 * Hc, Hc);
  gru_gate<<<Bc, 256, 0, stream>>>(gi, gh, last_hidden + (size_t)Bc * Hc, h1, hid1);

  // 4. attention scores + log_softmax (weights ARE the log-softmax, faithful)
  attn_scores<<<dim3(Lc / 8, Bc), 256, 0, stream>>>(h1, enc, logp);
  log_softmax_k<<<Bc, 256, 0, stream>>>(logp, attn);

  // 5. context
  context_k<<<Bc, 1024, 0, stream>>>(logp, enc, ctx);

  // 6. concat + tanh projection
  concat_build<<<Bc, 256, 0, stream>>>(h1, ctx, cc);
  wmma_gemm_nt<1><<<(Hc + 63) / 64, gemm_threads, 0, stream>>>(cc, concat_W, concat_b, cout, Hc, 2 * Hc);

  // 7. vocab projection (dominant GEMM: out_W streamed exactly once)
  wmma_gemm_nt<0><<<(Vc + 63) / 64, gemm_threads, 0, stream>>>(cout, out_W, out_b, out, Vc, Hc);
}